// FbpLayer_20418274525760
// MI455X (gfx1250) — compile-verified
//
#include <hip/hip_runtime.h>

typedef float v2f __attribute__((ext_vector_type(2)));
typedef float v8f __attribute__((ext_vector_type(8)));
typedef int   v4i __attribute__((ext_vector_type(4)));
typedef float v4f __attribute__((ext_vector_type(4)));

#define NB     4
#define NANG   360
#define NDET   512
#define LTAPS  255
#define OUTW   256
#define NPIX   (OUTW * OUTW)      /* 65536  */
#define SINSZ  (NANG * NDET)      /* 184320 */
#define NROWS  (NB * NANG)        /* 1440   */
#define NNZ    24000000

// ---------------------------------------------------------------------------
// Kernel 1: FBP ramp-filter correlation as a banded-Toeplitz WMMA GEMM.
// out[w, r] = sum_{k=0..254} f[k] * x[r, w + k - 127]   (zeros outside)
// Block = 128 threads (4 waves). Each wave computes a 16(w) x 16(row) tile.
// Block covers 64 w-positions x 16 rows. Grid = (1440/16) * (512/64) = 720.
// Result stored batch-interleaved: sinf[(a*512 + w)*4 + b].
// ---------------------------------------------------------------------------
__global__ __launch_bounds__(128) void fbp_filter_kernel(
    const float* __restrict__ sin_fan,   // [NROWS, NDET], row = b*360 + a
    const float* __restrict__ filt,      // [255]
    float* __restrict__ sinf)            // [SINSZ * 4] interleaved
{
  __shared__ float fpad[288];            // zero-padded filter, fpad[16+j] = f[j]
  __shared__ float slab[16][321];        // 16 rows x 320 t-values (+1 pad vs 64 banks)

  const int tid  = threadIdx.x;
  const int wv   = tid >> 5;             // wave id 0..3
  const int lane = tid & 31;

  const int rb = (int)blockIdx.x / 8;    // row-group 0..89
  const int wb = ((int)blockIdx.x % 8) * 64;  // block w base
  const int r0 = rb * 16;

  // Zero-padded filter into LDS (tap index k maps to fpad[k + 16], k in [-15, 271])
  for (int i = tid; i < 288; i += 128) {
    int j = i - 16;
    fpad[i] = (j >= 0 && j < LTAPS) ? filt[j] : 0.0f;
  }
  // Input slab: t in [wb-127, wb+192], SAME padding with zeros.
  for (int i = tid; i < 16 * 320; i += 128) {
    int tl = i % 320;
    int r  = i / 320;
    int t  = wb - 127 + tl;
    slab[r][tl] = (t >= 0 && t < NDET) ? sin_fan[(r0 + r) * NDET + t] : 0.0f;
  }
  __syncthreads();

  const int m   = lane & 15;             // A: M = lane & 15 ; B/C/D: N = lane & 15
  const int off = (lane >> 4) << 1;      // K sub-offset: 0 (lanes 0-15) or 2 (lanes 16-31)

  v8f c = {0.f, 0.f, 0.f, 0.f, 0.f, 0.f, 0.f, 0.f};
  const int tbase = wv * 16;             // this wave's t offset inside the slab
  int fidx = 16 + off - m;               // fpad index of tap for kk = 0, K = off

  #pragma unroll 4
  for (int kk = 0; kk < 68; ++kk) {      // K = 272 = 68 chunks of 4
    v2f a, b;
    a.x = fpad[fidx];                    // A[m, K=off]   = f[4kk + off     - m]
    a.y = fpad[fidx + 1];                // A[m, K=off+1] = f[4kk + off + 1 - m]
    const int tl = tbase + 4 * kk + off;
    b.x = slab[m][tl];                   // B[K=off,   n] = x[n, t]
    b.y = slab[m][tl + 1];               // B[K=off+1, n]
    c = __builtin_amdgcn_wmma_f32_16x16x4_f32(
        /*neg_a=*/false, a, /*neg_b=*/false, b,
        /*c_mod=*/(short)0, c, /*reuse_a=*/false, /*reuse_b=*/false);
    fidx += 4;
  }

  // C/D layout: VGPR j holds M = j (lanes 0-15) or M = j + 8 (lanes 16-31), N = lane & 15.
  const int row = r0 + m;                // 0..1439 = b*360 + a
  const int bat = row / NANG;
  const int ang = row % NANG;
  const int mhi = (lane >> 4) * 8;
  #pragma unroll
  for (int j = 0; j < 8; ++j) {
    const int w = wb + wv * 16 + mhi + j;
    sinf[((ang * NDET + w) << 2) + bat] = c[j];
  }
}

// ---------------------------------------------------------------------------
// Kernel 2: zero the interleaved accumulator (graph-replay safe).
// ---------------------------------------------------------------------------
__global__ __launch_bounds__(256) void fbp_zero_kernel(float* __restrict__ p, int n)
{
  int i = blockIdx.x * blockDim.x + threadIdx.x;
  if (i < n) p[i] = 0.0f;
}

// ---------------------------------------------------------------------------
// Kernel 3: sparse backprojection. Streams 288 MB of triplets with
// non-temporal b128 loads; one b128 gather per entry (4 batches interleaved);
// 4 f32 atomics per entry into one cache line of the interleaved accumulator.
// ---------------------------------------------------------------------------
__global__ __launch_bounds__(256) void fbp_scatter_kernel(
    const v4i* __restrict__ rows4,
    const v4i* __restrict__ cols4,
    const v4f* __restrict__ vals4,
    const float* __restrict__ sinf,      // [SINSZ * 4] interleaved, L2-resident
    float* __restrict__ acc,             // [NPIX * 4] interleaved, L2-resident
    int n4)
{
  const int stride = gridDim.x * blockDim.x;
  for (int i = blockIdx.x * blockDim.x + threadIdx.x; i < n4; i += stride) {
    v4i r = __builtin_nontemporal_load(&rows4[i]);
    v4i q = __builtin_nontemporal_load(&cols4[i]);
    v4f v = __builtin_nontemporal_load(&vals4[i]);
    #pragma unroll
    for (int j = 0; j < 4; ++j) {
      const float4 s = *(const float4*)(sinf + ((size_t)(unsigned)r[j] << 2));
      float*       a = acc + ((size_t)(unsigned)q[j] << 2);
      const float  w = v[j];
      atomicAdd(a + 0, w * s.x);
      atomicAdd(a + 1, w * s.y);
      atomicAdd(a + 2, w * s.z);
      atomicAdd(a + 3, w * s.w);
    }
  }
}

// ---------------------------------------------------------------------------
// Kernel 4: out[b*65536 + c] = acc[c*4 + b] * scale + bias
// ---------------------------------------------------------------------------
__global__ __launch_bounds__(256) void fbp_finalize_kernel(
    const float* __restrict__ acc,
    const float* __restrict__ scale,
    const float* __restrict__ bias,
    float* __restrict__ out)
{
  int i = blockIdx.x * blockDim.x + threadIdx.x;      // 0 .. NB*NPIX-1
  int c = i & (NPIX - 1);
  int b = i >> 16;
  out[i] = acc[(c << 2) + b] * scale[0] + bias[0];
}

// ---------------------------------------------------------------------------
extern "C" void kernel_launch(void* const* d_in, const int* in_sizes, int n_in,
                              void* d_out, int out_size, void* d_ws, size_t ws_size,
                              hipStream_t stream)
{
  (void)in_sizes; (void)n_in; (void)out_size; (void)ws_size;

  const float* sin_fan = (const float*)d_in[0];   // [4,360,512,1] f32
  const float* filt    = (const float*)d_in[1];   // [255,1,1]     f32
  const float* at_vals = (const float*)d_in[2];   // [NNZ]         f32
  const float* scale   = (const float*)d_in[3];   // [1]           f32
  const float* bias    = (const float*)d_in[4];   // [1]           f32
  const int*   at_rows = (const int*)d_in[5];     // [NNZ]         i32
  const int*   at_cols = (const int*)d_in[6];     // [NNZ]         i32
  float*       out     = (float*)d_out;           // [4,256,256,1] f32

  // Workspace layout: [0, SINSZ*4) interleaved filtered sinogram (2.95 MB),
  // then [SINSZ*4, SINSZ*4 + NPIX*4) interleaved accumulator (1 MB).
  float* sinf = (float*)d_ws;
  float* acc  = sinf + (size_t)SINSZ * 4;

  fbp_filter_kernel<<<720, 128, 0, stream>>>(sin_fan, filt, sinf);

  fbp_zero_kernel<<<(NPIX * NB + 255) / 256, 256, 0, stream>>>(acc, NPIX * NB);

  fbp_scatter_kernel<<<8192, 256, 0, stream>>>(
      (const v4i*)at_rows, (const v4i*)at_cols, (const v4f*)at_vals,
      sinf, acc, NNZ / 4);

  fbp_finalize_kernel<<<(NPIX * NB + 255) / 256, 256, 0, stream>>>(
      acc, scale, bias, out);
}